// BaseNetEmbedding_22411139351168
// MI455X (gfx1250) — compile-verified
//
#include <hip/hip_runtime.h>
#include <hip/hip_bf16.h>

#define BATCH 128
#define SEQ   2000
#define EMB   50
#define OUT   2
#define KPAD  52   // EMB padded so kb+1 (max 51) is always in-bounds

typedef __attribute__((ext_vector_type(2))) float v2f;
typedef __attribute__((ext_vector_type(8))) float v8f;

// ---------------------------------------------------------------------------
// Kernel 1: masked gather + partial sum pooling.
// grid = BATCH * split blocks, 256 threads (8 wave32) per block.
// Block (row, s) sums tokens [s*tpb, (s+1)*tpb) of `row` into
// partial[s][row][0..49]. Lanes 0..24 of a wave each own one float2 (pair of
// embedding dims); a wave processes every 8th token of the block's range.
// Deterministic LDS serial sum across the 8 waves (no float atomics).
// ---------------------------------------------------------------------------
__global__ __launch_bounds__(256)
void gather_pool_kernel(const int* __restrict__ t,
                        const float* __restrict__ emb,
                        float* __restrict__ partial,
                        int split) {
    const int row  = blockIdx.x / split;
    const int s    = blockIdx.x - row * split;
    const int tpb  = SEQ / split;          // split divides 2000 (8/4/2/1)
    const int wave = threadIdx.x >> 5;
    const int lane = threadIdx.x & 31;

    __shared__ float lds_part[8][EMB];

    const int* __restrict__ trow = t + (size_t)row * SEQ + (size_t)s * tpb;

    float ax = 0.f, ay = 0.f;
    if (lane < 25) {
        const size_t pair = (size_t)(2 * lane);
        // unroll x2 over this wave's tokens (stride 8 per wave, 8 waves)
        for (int tok = wave; tok < tpb; tok += 16) {
            const int i0 = trow[tok];
            const int t1 = tok + 8;
            const int i1 = (t1 < tpb) ? trow[t1] : -1;

            // prefetch one unrolled iteration ahead (gfx1250 global_prefetch)
            const int tp = tok + 16;
            if (tp < tpb) {
                const int ip = trow[tp];
                if (ip >= 0)
                    __builtin_prefetch(emb + (size_t)ip * EMB + pair, 0, 0);
            }

            float v0x = 0.f, v0y = 0.f, v1x = 0.f, v1y = 0.f;
            if (i0 >= 0) {   // wave-uniform branch (same token per wave)
                const float2 v = *(const float2*)(emb + (size_t)i0 * EMB + pair);
                v0x = v.x; v0y = v.y;
            }
            if (i1 >= 0) {
                const float2 v = *(const float2*)(emb + (size_t)i1 * EMB + pair);
                v1x = v.x; v1y = v.y;
            }
            ax += v0x + v1x;
            ay += v0y + v1y;
        }
        lds_part[wave][2 * lane]     = ax;
        lds_part[wave][2 * lane + 1] = ay;
    }
    __syncthreads();

    if (threadIdx.x < EMB) {
        float sum = 0.f;
        #pragma unroll
        for (int w = 0; w < 8; ++w) sum += lds_part[w][threadIdx.x];
        partial[((size_t)s * BATCH + row) * EMB + threadIdx.x] = sum;
    }
}

// ---------------------------------------------------------------------------
// Kernel 2: reduce partials -> pooled (LDS, K-padded), stage zero-padded
// B = W.T in LDS, then pooled @ W.T + b with ReLU via V_WMMA_F32_16X16X4_F32.
// One block, 8 waves; wave w owns rows [16w, 16w+16).
// K = 50 -> 13 WMMA steps of K=4; padding makes all operand loads
// unconditional ds_load_b64 (no exec-mask branches in the WMMA loop).
// ---------------------------------------------------------------------------
__global__ __launch_bounds__(256)
void pool_matmul_kernel(const float* __restrict__ partial,
                        const float* __restrict__ W,
                        const float* __restrict__ bias,
                        float* __restrict__ out,
                        int split) {
    __shared__ __align__(16) float pooled[BATCH][KPAD]; // 26.6 KB
    __shared__ __align__(16) float Bs[16][KPAD];        //  3.3 KB, B[k][n]=W[n][k]

    // Phase 1a: zero the B staging tile (covers N in [2,16) and K tail).
    for (int i = threadIdx.x; i < 16 * KPAD; i += 256)
        (&Bs[0][0])[i] = 0.f;
    // Phase 1b: deterministic reduction over split partials + 1/BATCH scale.
    for (int i = threadIdx.x; i < BATCH * EMB; i += 256) {
        float sum = 0.f;
        for (int s = 0; s < split; ++s)
            sum += partial[(size_t)s * BATCH * EMB + i];
        const int m = i / EMB;
        const int k = i - m * EMB;
        pooled[m][k] = sum * (1.0f / (float)BATCH);
    }
    // Phase 1c: zero pooled K-padding.
    for (int i = threadIdx.x; i < BATCH * (KPAD - EMB); i += 256)
        pooled[i >> 1][EMB + (i & 1)] = 0.f;
    __syncthreads();
    // Phase 1d: fill valid B rows (after barrier not needed before, Bs rows
    // 0..1 only touched here; do before the sync instead)
    // -- moved above the barrier:
    // (kept separate loop for clarity; see below)

    // NOTE: W fill must be before __syncthreads; do it here via second barrier
    for (int i = threadIdx.x; i < OUT * EMB; i += 256) {
        const int n = i / EMB;
        const int k = i - n * EMB;
        Bs[n][k] = W[i];
    }
    __syncthreads();

    const int wave = threadIdx.x >> 5;
    const int lane = threadIdx.x & 31;
    const int nrow = lane & 15;   // M within A-tile, and N for B/C/D
    const int half = lane >> 4;   // 0: K pair {0,1}; 1: K pair {2,3}

    const int row0 = wave * 16;
    const float* __restrict__ arow = &pooled[row0 + nrow][0];
    const float* __restrict__ brow = &Bs[nrow][0];
    v8f acc = {};

    // 13 K-steps of 4 (uniform loop; EXEC all-ones at each WMMA).
    #pragma unroll
    for (int k0 = 0; k0 < EMB; k0 += 4) {
        const int kb = k0 + 2 * half;                 // even -> 8B aligned
        const v2f a  = *(const v2f*)(arow + kb);      // ds_load_b64
        const v2f bm = *(const v2f*)(brow + kb);      // ds_load_b64
        acc = __builtin_amdgcn_wmma_f32_16x16x4_f32(
            /*neg_a=*/false, a, /*neg_b=*/false, bm,
            /*c_mod=*/(short)0, acc, /*reuse_a=*/false, /*reuse_b=*/false);
    }

    // D layout: element (M = v + 8*half, N = lane%16) in acc[v]
    if (nrow < OUT) {
        const float bv = bias[nrow];
        #pragma unroll
        for (int v = 0; v < 8; ++v) {
            const int m = row0 + v + 8 * half;
            float r = acc[v] + bv;
            out[(size_t)m * OUT + nrow] = (r > 0.f) ? r : 0.f;
        }
    }
}

extern "C" void kernel_launch(void* const* d_in, const int* in_sizes, int n_in,
                              void* d_out, int out_size, void* d_ws, size_t ws_size,
                              hipStream_t stream) {
    const int*   t    = (const int*)  d_in[0];   // [128, 2000]
    const float* emb  = (const float*)d_in[1];   // [1e6, 50]
    const float* W    = (const float*)d_in[2];   // [2, 50]
    const float* bias = (const float*)d_in[3];   // [2]
    float*       out  = (float*)d_out;           // [128, 2]
    float*       part = (float*)d_ws;            // [split, 128, 50]

    int split = 8;                                // all of 8/4/2/1 divide 2000
    while (split > 1 &&
           (size_t)split * BATCH * EMB * sizeof(float) > ws_size)
        split >>= 1;

    gather_pool_kernel<<<dim3(BATCH * split), dim3(256), 0, stream>>>(
        t, emb, part, split);
    pool_matmul_kernel<<<dim3(1), dim3(256), 0, stream>>>(
        part, W, bias, out, split);
}